// CHSLoss_75582834475514
// MI455X (gfx1250) — compile-verified
//
#include <hip/hip_runtime.h>
#include <hip/hip_bf16.h>

typedef __attribute__((ext_vector_type(2))) float v2f;
typedef __attribute__((ext_vector_type(8))) float v8f;
typedef __attribute__((__vector_size__(4 * sizeof(int)))) int v4i_vs;

#define BATCH 16
#define H 192
#define W 256
#define GH 1536
#define GW 2048
#define NPB (H * W)          // 49152 elements per batch row
#define NTOT (BATCH * NPB)   // 786432
#define NPART 1024

// workspace layout (floats)
#define WS_DMAP 0
#define WS_THR  (NTOT)        // 32 floats
#define WS_PART (NTOT + 32)   // NPART floats

// ---------------- async-to-LDS helpers (CDNA5 path, guarded) ----------------
__device__ __forceinline__ void async_load_b128(const float* g, float* l) {
#if __has_builtin(__builtin_amdgcn_global_load_async_to_lds_b128)
    __builtin_amdgcn_global_load_async_to_lds_b128(
        (__attribute__((address_space(1))) v4i_vs*)g,
        (__attribute__((address_space(3))) v4i_vs*)l,
        0, 0);
#else
    *(float4*)l = *(const float4*)g;   // fallback: direct copy through VGPRs
#endif
}

__device__ __forceinline__ void wait_async() {
#if __has_builtin(__builtin_amdgcn_s_wait_asynccnt)
    __builtin_amdgcn_s_wait_asynccnt(0);
#elif __has_builtin(__builtin_amdgcn_global_load_async_to_lds_b128)
    asm volatile("s_wait_asynccnt 0" ::: "memory");
#endif
}

// ---------------------------------------------------------------------------
// Kernel 1: 8x8 block-sum pooling of gt_density -> dmap_gt.
// Grid: (strip=8, iblk=12, b=16), block = 128 threads (4 waves).
// Each WG: gt rows [iblk*128, +128), cols [strip*256, +256) -> out 16x32 tile.
// 4 chunks of 32 rows: async-load 32x256 f32 slab to LDS, horizontal 8-sum to
// tmp[32][32], then waves 0..1 do the vertical 8-sum as WMMA f32 16x16x4 with
// a 0/1 pooling A-matrix (K accumulates to 128 across chunks).
// ---------------------------------------------------------------------------
#define K1_THREADS 128
#define CHROWS 32
#define SLABC 256
#define TMPS 33   // pad to break LDS bank conflicts on B reads

__global__ __launch_bounds__(K1_THREADS)
void pool_kernel(const float* __restrict__ gt, float* __restrict__ dmap)
{
    __shared__ __align__(16) float slab[CHROWS * SLABC]; // 32 KB
    __shared__ float tmp[CHROWS * TMPS];                 // 4.2 KB

    const int strip = blockIdx.x;   // 0..7   (256 gt cols = 32 out cols)
    const int iblk  = blockIdx.y;   // 0..11  (128 gt rows = 16 out rows)
    const int b     = blockIdx.z;   // 0..15
    const int tid   = threadIdx.x;
    const int lane  = tid & 31;
    const int wave  = tid >> 5;

    const size_t gbase0 =
        ((size_t)b * GH + (size_t)iblk * 128) * GW + (size_t)strip * 256;

    v8f acc = {};          // 16x16 f32 accumulator tile (waves 0,1)
    const int ntile = wave;
    const int mrow  = lane & 15;          // A: M index / B,D: N index
    const int koff  = (lane >> 4) << 1;   // K offset 0 or 2 per half-wave

    for (int ch = 0; ch < 4; ++ch) {
        const size_t gb = gbase0 + (size_t)(ch * CHROWS) * GW;
        // 32 rows * 64 float4 = 2048 b128 transfers; 16 per thread, coalesced.
        #pragma unroll 4
        for (int it = 0; it < 16; ++it) {
            int idx = it * K1_THREADS + tid;  // 0..2047
            int r   = idx >> 6;
            int c4  = idx & 63;
            async_load_b128(gt + gb + (size_t)r * GW + (size_t)(c4 * 4),
                            &slab[r * SLABC + c4 * 4]);
        }
        wait_async();
        __syncthreads();

        // horizontal 8-sum: tmp[r][jo] = sum_q slab[r][8*jo+q]
        for (int t = tid; t < CHROWS * 32; t += K1_THREADS) {
            int r = t >> 5, jo = t & 31;
            const float4* p = (const float4*)&slab[r * SLABC + jo * 8];
            float4 x = p[0], y = p[1];
            tmp[r * TMPS + jo] = ((x.x + x.y) + (x.z + x.w))
                               + ((y.x + y.y) + (y.z + y.w));
        }
        __syncthreads();

        // vertical 8-sum via matrix pipe: acc += P(16x32chunk) * tmp(32x32)
        if (wave < 2) {
            const int col = ntile * 16 + mrow;   // B/N column within tmp
            #pragma unroll
            for (int k0 = 0; k0 < CHROWS; k0 += 4) {
                int ka = ch * CHROWS + k0 + koff; // absolute K (0..127)
                int kb = k0 + koff;               // K within chunk
                v2f a, bv;
                a.x = (( ka      >> 3) == mrow) ? 1.0f : 0.0f;
                a.y = (((ka + 1) >> 3) == mrow) ? 1.0f : 0.0f;
                bv.x = tmp[ kb      * TMPS + col];
                bv.y = tmp[(kb + 1) * TMPS + col];
                acc = __builtin_amdgcn_wmma_f32_16x16x4_f32(
                        false, a, false, bv, (short)0, acc, false, false);
            }
        }
        __syncthreads();
    }

    if (wave < 2) {
        // D layout: VGPR r -> M=r (lanes 0-15) / M=8+r (lanes 16-31), N=lane&15
        const int mh  = (lane >> 4) * 8;
        const int col = strip * 32 + ntile * 16 + mrow;
        float* dst = dmap + (size_t)b * NPB;
        #pragma unroll
        for (int r = 0; r < 8; ++r) {
            int row = iblk * 16 + mh + r;
            dst[row * W + col] = acc[r];
        }
    }
}

// ---------------------------------------------------------------------------
// Kernel 2: exact k-th largest |dmap_gt - map| per (map, batch) row via
// 4-pass MSB radix select on float bit patterns (monotone for err >= 0).
// Grid: 32 blocks (row = mi*16 + b), 256 threads, LDS histogram (ds_add).
// ---------------------------------------------------------------------------
__global__ __launch_bounds__(256)
void select_kernel(const float* __restrict__ map0, const float* __restrict__ map1,
                   const float* __restrict__ dmap, const float* __restrict__ processp,
                   float* __restrict__ thr)
{
    __shared__ unsigned int hist[256];
    __shared__ unsigned int s_prefix, s_want;

    const int row = blockIdx.x;      // 0..31
    const int mi  = row >> 4;
    const int b   = row & 15;
    const int tid = threadIdx.x;

    const float process = processp[0];
    const int num = (int)((float)NPB * 0.1f * process);
    if (num < 1) {
        // reference falls back to plain (m - gt)^2; +inf threshold reproduces it
        if (tid == 0) thr[row] = __builtin_inff();
        return;
    }

    const float* m = ((mi == 0) ? map0 : map1) + (size_t)b * NPB;
    const float* g = dmap + (size_t)b * NPB;

    if (tid == 0) { s_prefix = 0u; s_want = (unsigned)num; }
    __syncthreads();

    for (int shift = 24; shift >= 0; shift -= 8) {
        hist[tid] = 0u;
        __syncthreads();
        const unsigned prefix  = s_prefix;
        const unsigned mask_hi = (shift == 24) ? 0u : (0xFFFFFFFFu << (shift + 8));
        for (int e = tid; e < NPB; e += 256) {
            unsigned u = __float_as_uint(fabsf(g[e] - m[e]));
            if ((u & mask_hi) == prefix)
                atomicAdd(&hist[(u >> shift) & 255u], 1u);
        }
        __syncthreads();
        if (tid == 0) {
            unsigned want = s_want, cum = 0u;
            for (int d = 255; d >= 0; --d) {
                cum += hist[d];
                if (cum >= want) {
                    s_prefix = prefix | ((unsigned)d << shift);
                    s_want   = want - (cum - hist[d]);
                    break;
                }
            }
        }
        __syncthreads();
    }
    if (tid == 0) thr[row] = __uint_as_float(s_prefix);
}

// ---------------------------------------------------------------------------
// Kernel 3: fused masked squared-error, fixed-order block partial sums.
// ---------------------------------------------------------------------------
__global__ __launch_bounds__(256)
void loss_kernel(const float* __restrict__ map0, const float* __restrict__ map1,
                 const float* __restrict__ dmap, const float* __restrict__ thr,
                 const float* __restrict__ processp, float* __restrict__ partial)
{
    __shared__ float red[256];
    const float w  = 1.0f * processp[0];
    const int  tid = threadIdx.x;

    float acc = 0.0f;
    for (size_t e = (size_t)blockIdx.x * 256 + tid; e < (size_t)NTOT;
         e += (size_t)gridDim.x * 256) {
        const int b = (int)(e / NPB);
        const float g  = dmap[e];
        const float m0 = map0[e], m1 = map1[e];
        const float t0 = thr[b], t1 = thr[16 + b];
        const float e0 = fabsf(g - m0), e1 = fabsf(g - m1);
        const float c0 = w * m0 + (1.0f - w) * g;
        const float c1 = w * m1 + (1.0f - w) * g;
        const float s0 = (e0 >= t0) ? c1 : g;   // i=0 uses comb of map1
        const float s1 = (e1 >= t1) ? c0 : g;   // i=1 uses comb of map0
        const float d0 = m0 - s0, d1 = m1 - s1;
        acc += d0 * d0 + d1 * d1;
    }
    red[tid] = acc;
    __syncthreads();
    for (int s = 128; s > 0; s >>= 1) {
        if (tid < s) red[tid] += red[tid + s];
        __syncthreads();
    }
    if (tid == 0) partial[blockIdx.x] = red[0];
}

__global__ __launch_bounds__(256)
void final_kernel(const float* __restrict__ partial, float* __restrict__ out)
{
    __shared__ float red[256];
    const int tid = threadIdx.x;
    float acc = 0.0f;
    for (int i = tid; i < NPART; i += 256) acc += partial[i];
    red[tid] = acc;
    __syncthreads();
    for (int s = 128; s > 0; s >>= 1) {
        if (tid < s) red[tid] += red[tid + s];
        __syncthreads();
    }
    if (tid == 0) out[0] = red[0];
}

// ---------------------------------------------------------------------------
extern "C" void kernel_launch(void* const* d_in, const int* in_sizes, int n_in,
                              void* d_out, int out_size, void* d_ws, size_t ws_size,
                              hipStream_t stream)
{
    const float* map0 = (const float*)d_in[0];
    const float* map1 = (const float*)d_in[1];
    const float* gt   = (const float*)d_in[2];
    const float* proc = (const float*)d_in[3];

    float* ws   = (float*)d_ws;
    float* dmap = ws + WS_DMAP;
    float* thr  = ws + WS_THR;
    float* part = ws + WS_PART;
    float* out  = (float*)d_out;

    dim3 g1(GW / 256, H / 16, BATCH);   // (8, 12, 16)
    pool_kernel<<<g1, K1_THREADS, 0, stream>>>(gt, dmap);
    select_kernel<<<32, 256, 0, stream>>>(map0, map1, dmap, proc, thr);
    loss_kernel<<<NPART, 256, 0, stream>>>(map0, map1, dmap, thr, proc, part);
    final_kernel<<<1, 256, 0, stream>>>(part, out);
}